// RKNCell_12206297055879
// MI455X (gfx1250) — compile-verified
//
#include <hip/hip_runtime.h>
#include <hip/hip_bf16.h>

// ---------------------------------------------------------------------------
// RKN cell forward, fused for MI455X (gfx1250, wave32).
// Memory-bound problem (~11 MB traffic @ 23.3 TB/s); we stream every output
// once, never materialize the banded 256x256 transition matrices, and use
// V_WMMA_F32_16X16X4_F32 (full fp32 precision) for the MLP GEMM.
// ---------------------------------------------------------------------------

#define B_   512
#define LOD_ 256
#define LSD_ 512
#define BW_  3
#define NB_  15
#define HID_ 128
#define NE_  1780   // number of band entries for 256x256, |i-j|<=3

// d_out float offsets (return order: pm, pcu, pcl, pcs, nm, ncu, ncl, ncs, kg)
#define OFF_PM   0
#define OFF_PCU  (OFF_PM  + B_*LSD_)   // 262144
#define OFF_PCL  (OFF_PCU + B_*LOD_)
#define OFF_PCS  (OFF_PCL + B_*LOD_)
#define OFF_NM   (OFF_PCS + B_*LOD_)
#define OFF_NCU  (OFF_NM  + B_*LSD_)
#define OFF_NCL  (OFF_NCU + B_*LOD_)
#define OFF_NCS  (OFF_NCL + B_*LOD_)
#define OFF_KG   (OFF_NCS + B_*LOD_)

typedef __attribute__((ext_vector_type(2))) float v2f;
typedef __attribute__((ext_vector_type(8))) float v8f;

// Flat index of the first band entry of row i (row-major scan of the mask).
__device__ __forceinline__ int band_offset(int i) {
    if (i < 3)        return (i * (i + 7)) >> 1;   // 0, 4, 9
    else if (i <= 253) return 15 + 7 * (i - 3);    // full 7-wide rows
    else if (i == 254) return 1771;
    else               return 1776;
}

// ---------------------------------------------------------------------------
// Kernel 1: Kalman observation update (elementwise).  1 thread = (b, i).
// ---------------------------------------------------------------------------
__global__ void k_kalman(const float* __restrict__ prior_mean,
                         const float* __restrict__ cov_u,
                         const float* __restrict__ cov_l,
                         const float* __restrict__ cov_s,
                         const float* __restrict__ obs,
                         const float* __restrict__ obs_var,
                         const int*  __restrict__ obs_valid,
                         float* __restrict__ out) {
    const int b = blockIdx.x;
    const int i = threadIdx.x;

    const float pmu = prior_mean[b * LSD_ + i];
    const float pml = prior_mean[b * LSD_ + LOD_ + i];
    const float cu  = cov_u[b * LOD_ + i];
    const float cl  = cov_l[b * LOD_ + i];
    const float cs  = cov_s[b * LOD_ + i];
    const float ob  = obs[b * LOD_ + i];
    const float ovr = obs_var[b * LOD_ + i];
    const bool  valid = obs_valid[b] != 0;

    const float denom = cu + ovr;
    const float qu = cu / denom;
    const float ql = cs / denom;
    const float resid = ob - pmu;
    const float cf = 1.0f - qu;

    const float post_u = valid ? fmaf(qu, resid, pmu) : pmu;
    const float post_l = valid ? fmaf(ql, resid, pml) : pml;
    const float pcu = valid ? cf * cu        : cu;
    const float pcl = valid ? fmaf(-ql, cs, cl) : cl;
    const float pcs = valid ? cf * cs        : cs;
    const float kgu = valid ? qu : 0.0f;
    const float kgl = valid ? ql : 0.0f;

    out[OFF_PM  + b * LSD_ + i]        = post_u;
    out[OFF_PM  + b * LSD_ + LOD_ + i] = post_l;
    out[OFF_PCU + b * LOD_ + i] = pcu;
    out[OFF_PCL + b * LOD_ + i] = pcl;
    out[OFF_PCS + b * LOD_ + i] = pcs;
    out[OFF_KG  + b * LSD_ + i]        = kgu;
    out[OFF_KG  + b * LSD_ + LOD_ + i] = kgl;
}

// ---------------------------------------------------------------------------
// Kernel 2: h = tanh(pm @ W1 + b1) via V_WMMA_F32_16X16X4_F32, then
//           coeff = softmax(h @ W2 + b2).  1 block = 16 batch rows,
//           8 waves (256 thr), each wave owns one 16-col tile of HID=128.
// ---------------------------------------------------------------------------
__global__ void k_mlp_wmma(const float* __restrict__ pm,   // d_out + OFF_PM
                           const float* __restrict__ W1,   // [LSD_][HID_]
                           const float* __restrict__ b1,   // [HID_]
                           const float* __restrict__ W2,   // [HID_][NB_]
                           const float* __restrict__ b2,   // [NB_]
                           float* __restrict__ coeff_out)  // [B_][NB_]
{
    __shared__ float hbuf[16][HID_];
    __shared__ float logits[16][NB_];

    const int t    = threadIdx.x;
    const int lane = t & 31;
    const int wave = t >> 5;         // 0..7
    const int half = lane >> 4;      // 0: K-pair {0,1}, 1: K-pair {2,3}
    const int l15  = lane & 15;
    const int m0   = blockIdx.x * 16;
    const int n0   = wave * 16;

    // A fragment source: this lane's batch row of post_mean
    const float* arow = pm + (m0 + l15) * LSD_;

    v8f c = {};
    #pragma unroll 8
    for (int k = 0; k < LSD_; k += 4) {
        // A: 16x4 fp32 tile. lanes 0-15: {K=k,k+1}, lanes 16-31: {K=k+2,k+3}
        v2f a = *(const v2f*)(arow + k + 2 * half);
        // B: 4x16 fp32 tile (rows of W1). lanes 0-15: rows k,k+1; 16-31: k+2,k+3
        v2f bf;
        bf.x = W1[(k + 2 * half)     * HID_ + n0 + l15];
        bf.y = W1[(k + 2 * half + 1) * HID_ + n0 + l15];
        c = __builtin_amdgcn_wmma_f32_16x16x4_f32(
                /*neg_a=*/false, a, /*neg_b=*/false, bf,
                /*c_mod=*/(short)0, c, /*reuse_a=*/false, /*reuse_b=*/false);
    }

    // C/D layout: VGPR r -> M = r + 8*half, N = l15.  Apply bias + tanh.
    const int col = n0 + l15;
    const float bias = b1[col];
    #pragma unroll
    for (int r = 0; r < 8; ++r) {
        hbuf[r + 8 * half][col] = tanhf(c[r] + bias);
    }
    __syncthreads();

    // logits = h @ W2 + b2  (16 rows x 15 coeffs = 240 dot products of 128)
    if (t < 16 * NB_) {
        const int row = t / NB_;
        const int j   = t % NB_;
        float acc = b2[j];
        #pragma unroll 4
        for (int k = 0; k < HID_; ++k)
            acc = fmaf(hbuf[row][k], W2[k * NB_ + j], acc);
        logits[row][j] = acc;
    }
    __syncthreads();

    // softmax over NB_=15 per row; 16 threads, one row each
    if (t < 16) {
        float mx = logits[t][0];
        #pragma unroll
        for (int j = 1; j < NB_; ++j) mx = fmaxf(mx, logits[t][j]);
        float e[NB_];
        float s = 0.0f;
        #pragma unroll
        for (int j = 0; j < NB_; ++j) { e[j] = __expf(logits[t][j] - mx); s += e[j]; }
        const float inv = 1.0f / s;
        #pragma unroll
        for (int j = 0; j < NB_; ++j)
            coeff_out[(m0 + t) * NB_ + j] = e[j] * inv;
    }
}

// ---------------------------------------------------------------------------
// Kernel 3: band transition.  1 block = 1 batch, 1 thread = 1 output row i.
// Expands the <=7-wide band of tm11/tm12/tm21/tm22 from coeff . basis
// (contiguous basis loads), then mean propagation + covariance propagation.
// ---------------------------------------------------------------------------
__global__ void k_band_trans(const float* __restrict__ out_ro,   // d_out (pm/pcu/pcl/pcs)
                             const float* __restrict__ coeff,    // [B_][NB_]
                             const float* __restrict__ b11,
                             const float* __restrict__ b12,
                             const float* __restrict__ b21,
                             const float* __restrict__ b22,
                             const float* __restrict__ log_noise, // [LSD_]
                             float* __restrict__ out)
{
    __shared__ float cf[NB_];
    __shared__ float mu_u[LOD_], mu_l[LOD_];
    __shared__ float su[LOD_], sl[LOD_], ss[LOD_];

    const int b = blockIdx.x;
    const int i = threadIdx.x;

    mu_u[i] = out_ro[OFF_PM  + b * LSD_ + i];
    mu_l[i] = out_ro[OFF_PM  + b * LSD_ + LOD_ + i];
    su[i]   = out_ro[OFF_PCU + b * LOD_ + i];
    sl[i]   = out_ro[OFF_PCL + b * LOD_ + i];
    ss[i]   = out_ro[OFF_PCS + b * LOD_ + i];
    if (i < NB_) cf[i] = coeff[b * NB_ + i];
    __syncthreads();

    const int jlo  = max(0, i - BW_);
    const int jhi  = min(LOD_ - 1, i + BW_);
    const int span = jhi - jlo + 1;
    const int e0   = band_offset(i);

    float a11[7], a12[7], a21[7], a22[7];
    #pragma unroll
    for (int d = 0; d < 7; ++d) { a11[d] = 0.f; a12[d] = 0.f; a21[d] = 0.f; a22[d] = 0.f; }

    // band coefficients: sum_n coeff[n] * basis[n][e0 + d]   (contiguous in d)
    for (int n = 0; n < NB_; ++n) {
        const float cn = cf[n];
        const float* p11 = b11 + n * NE_ + e0;
        const float* p12 = b12 + n * NE_ + e0;
        const float* p21 = b21 + n * NE_ + e0;
        const float* p22 = b22 + n * NE_ + e0;
        #pragma unroll
        for (int d = 0; d < 7; ++d) {
            if (d < span) {
                a11[d] = fmaf(cn, p11[d], a11[d]);
                a12[d] = fmaf(cn, p12[d], a12[d]);
                a21[d] = fmaf(cn, p21[d], a21[d]);
                a22[d] = fmaf(cn, p22[d], a22[d]);
            }
        }
    }
    // unit diagonal on tm11, tm22
    const int dd = i - jlo;
    #pragma unroll
    for (int d = 0; d < 7; ++d) {
        if (d == dd) { a11[d] += 1.0f; a22[d] += 1.0f; }
    }

    float nmu = 0.f, nml = 0.f, ncu = 0.f, ncl = 0.f, ncs = 0.f;
    #pragma unroll
    for (int d = 0; d < 7; ++d) {
        if (d < span) {
            const int j = jlo + d;
            const float muu = mu_u[j], mul = mu_l[j];
            const float du = su[j], dl = sl[j], ds = ss[j];
            const float t11 = a11[d], t12 = a12[d], t21 = a21[d], t22 = a22[d];
            nmu = fmaf(t11, muu, fmaf(t12, mul, nmu));
            nml = fmaf(t21, muu, fmaf(t22, mul, nml));
            ncu += t11 * t11 * du + 2.0f * t11 * t12 * ds + t12 * t12 * dl;
            ncl += t21 * t21 * du + 2.0f * t21 * t22 * ds + t22 * t22 * dl;
            ncs += t21 * t11 * du + (t22 * t11 + t21 * t12) * ds + t22 * t12 * dl;
        }
    }

    // softplus transition noise (log_noise is (1, LSD), broadcast over batch)
    const float tcu = log1pf(__expf(log_noise[i]));
    const float tcl = log1pf(__expf(log_noise[LOD_ + i]));

    out[OFF_NM  + b * LSD_ + i]        = nmu;
    out[OFF_NM  + b * LSD_ + LOD_ + i] = nml;
    out[OFF_NCU + b * LOD_ + i] = ncu + tcu;
    out[OFF_NCL + b * LOD_ + i] = ncl + tcl;
    out[OFF_NCS + b * LOD_ + i] = ncs;
}

// ---------------------------------------------------------------------------
extern "C" void kernel_launch(void* const* d_in, const int* in_sizes, int n_in,
                              void* d_out, int out_size, void* d_ws, size_t ws_size,
                              hipStream_t stream) {
    const float* prior_mean = (const float*)d_in[0];
    const float* cov_u      = (const float*)d_in[1];
    const float* cov_l      = (const float*)d_in[2];
    const float* cov_s      = (const float*)d_in[3];
    const float* obs        = (const float*)d_in[4];
    const float* obs_var    = (const float*)d_in[5];
    const int*   obs_valid  = (const int*)  d_in[6];
    const float* W1         = (const float*)d_in[7];
    const float* b1         = (const float*)d_in[8];
    const float* W2         = (const float*)d_in[9];
    const float* b2         = (const float*)d_in[10];
    const float* tm11_basis = (const float*)d_in[11];
    const float* tm12_basis = (const float*)d_in[12];
    const float* tm21_basis = (const float*)d_in[13];
    const float* tm22_basis = (const float*)d_in[14];
    const float* log_noise  = (const float*)d_in[15];

    float* out   = (float*)d_out;
    float* coeff = (float*)d_ws;   // B_*NB_ floats = 30 KB scratch

    k_kalman<<<B_, LOD_, 0, stream>>>(prior_mean, cov_u, cov_l, cov_s,
                                      obs, obs_var, obs_valid, out);

    k_mlp_wmma<<<B_ / 16, 256, 0, stream>>>(out + OFF_PM, W1, b1, W2, b2, coeff);

    k_band_trans<<<B_, LOD_, 0, stream>>>(out, coeff,
                                          tm11_basis, tm12_basis, tm21_basis, tm22_basis,
                                          log_noise, out);
}